// GNNNet_67577015435759
// MI455X (gfx1250) — compile-verified
//
#include <hip/hip_runtime.h>
#include <cstdint>

typedef float v2f __attribute__((ext_vector_type(2)));
typedef float v8f __attribute__((ext_vector_type(8)));

__device__ __forceinline__ void atomic_fadd(float* p, float v) {
    __hip_atomic_fetch_add(p, v, __ATOMIC_RELAXED, __HIP_MEMORY_SCOPE_AGENT);
}

// ---------------- zero scratch ----------------
__global__ void k_zero(float* p, long long n) {
    long long i = (long long)blockIdx.x * blockDim.x + threadIdx.x;
    if (i < n) p[i] = 0.0f;
}

// ---------------- degree (u32 atomic) ----------------
__global__ void k_deg(const int* __restrict__ dst, unsigned* __restrict__ deg, long long E) {
    long long e = (long long)blockIdx.x * blockDim.x + threadIdx.x;
    if (e >= E) return;
    __hip_atomic_fetch_add(&deg[dst[e]], 1u, __ATOMIC_RELAXED, __HIP_MEMORY_SCOPE_AGENT);
}

// ---------------- deg -> rsqrt(deg+1), in place ----------------
__global__ void k_dinv(void* degdinv, int n) {
    int i = blockIdx.x * blockDim.x + threadIdx.x;
    if (i >= n) return;
    unsigned c = ((unsigned*)degdinv)[i];
    ((float*)degdinv)[i] = rsqrtf((float)c + 1.0f);
}

// ---------------- per-channel sum / sumsq over nodes ----------------
template <int C>
__global__ void k_stats(const float* __restrict__ Z, int stride, int n,
                        float* __restrict__ gsum, float* __restrict__ gss) {
    __shared__ float ssum[256];
    __shared__ float sss[256];
    const int G = 256 / C;
    int c = threadIdx.x % C;
    int g = threadIdx.x / C;
    float s = 0.0f, q = 0.0f;
    for (int r = blockIdx.x * G + g; r < n; r += gridDim.x * G) {
        float v = Z[r * stride + c];
        s += v; q += v * v;
    }
    ssum[threadIdx.x] = s; sss[threadIdx.x] = q;
    __syncthreads();
    for (int off = 128; off >= C; off >>= 1) {
        if ((int)threadIdx.x < off) {
            ssum[threadIdx.x] += ssum[threadIdx.x + off];
            sss[threadIdx.x]  += sss[threadIdx.x + off];
        }
        __syncthreads();
    }
    if ((int)threadIdx.x < C) {
        atomic_fadd(&gsum[threadIdx.x], ssum[threadIdx.x]);
        atomic_fadd(&gss[threadIdx.x],  sss[threadIdx.x]);
    }
}

// ---------------- BN affine: a = g*rsqrt(var+eps), s = b - mu*a ----------------
__global__ void k_finalize(const float* __restrict__ sum, const float* __restrict__ ss,
                           const float* __restrict__ g, const float* __restrict__ bb,
                           float* __restrict__ a, float* __restrict__ sft, float invN) {
    int c = threadIdx.x;
    float mu  = sum[c] * invN;
    float var = ss[c] * invN - mu * mu;
    float inv = rsqrtf(var + 1e-5f);
    float ac  = g[c] * inv;
    a[c] = ac;
    sft[c] = bb[c] - mu * ac;
}

// ---------------- layer 1 transform: bn0(x) * W1 (1x8), no relu ----------------
__global__ void k_h1(const float* __restrict__ x, const float* __restrict__ W1,
                     const float* __restrict__ dinv, const float* __restrict__ a0s,
                     float* __restrict__ F, float* __restrict__ H, int n) {
    int i = blockIdx.x * blockDim.x + threadIdx.x;
    if (i >= n) return;
    float xn = x[i] * a0s[0] + a0s[1];
    float d  = dinv[i];
    float d2 = d * d;
    #pragma unroll
    for (int c = 0; c < 8; ++c) {
        float h = xn * W1[c];
        H[i * 32 + c] = h;
        F[i * 32 + c] = h * d2;   // self-loop init for agg
    }
}

// ---------------- generic transform: relu(a*F+s) @ W, self-loop init ----------------
template <int CIN, int COUT>
__global__ void k_transform(float* __restrict__ F, float* __restrict__ H,
                            const float* __restrict__ dinv,
                            const float* __restrict__ a, const float* __restrict__ s,
                            const float* __restrict__ W, int n) {
    __shared__ float w[CIN * COUT];
    for (int t = threadIdx.x; t < CIN * COUT; t += blockDim.x) w[t] = W[t];
    __syncthreads();
    int i = blockIdx.x * blockDim.x + threadIdx.x;
    if (i >= n) return;
    float in[CIN];
    #pragma unroll
    for (int c = 0; c < CIN; ++c) {
        float v = F[i * 32 + c] * a[c] + s[c];
        in[c] = v > 0.0f ? v : 0.0f;
    }
    float d  = dinv[i];
    float d2 = d * d;
    #pragma unroll
    for (int co = 0; co < COUT; ++co) {
        float acc = 0.0f;
        #pragma unroll
        for (int ci = 0; ci < CIN; ++ci) acc += in[ci] * w[ci * COUT + co];
        H[i * 32 + co] = acc;
        F[i * 32 + co] = acc * d2;  // self-loop init for agg
    }
}

// ---------------- edge scatter: F[dst] += H[src] * dinv[src]*dinv[dst] ----------------
template <int C>
__global__ void k_edge(const int* __restrict__ src, const int* __restrict__ dst,
                       const float* __restrict__ dinv, const float* __restrict__ H,
                       float* __restrict__ F, long long E) {
    const int Q = C / 4;
    long long tid = (long long)blockIdx.x * blockDim.x + threadIdx.x;
    long long e = tid / Q;
    int q = (int)(tid % Q);
    if (e >= E) return;
    int sN = src[e], dN = dst[e];
    float coef = dinv[sN] * dinv[dN];
    const float4 hv = *(const float4*)(H + (size_t)sN * 32 + q * 4);
    float* fp = F + (size_t)dN * 32 + q * 4;
    atomic_fadd(fp + 0, hv.x * coef);
    atomic_fadd(fp + 1, hv.y * coef);
    atomic_fadd(fp + 2, hv.z * coef);
    atomic_fadd(fp + 3, hv.w * coef);
}

// ---------------- pool: relu(a*F+s) then segment-max (uint bits, values >= 0) ----------------
__global__ void k_pool(const float* __restrict__ F, const int* __restrict__ batch,
                       const float* __restrict__ a, const float* __restrict__ s,
                       float* __restrict__ pooled, int n) {
    int i = blockIdx.x * blockDim.x + threadIdx.x;
    if (i >= n) return;
    int b = batch[i];
    unsigned* pb = (unsigned*)(pooled + (size_t)b * 32);
    #pragma unroll
    for (int c = 0; c < 32; ++c) {
        float v = F[i * 32 + c] * a[c] + s[c];
        v = v > 0.0f ? v : 0.0f;
        __hip_atomic_fetch_max(&pb[c], __float_as_uint(v),
                               __ATOMIC_RELAXED, __HIP_MEMORY_SCOPE_AGENT);
    }
}

// ---------------- MLP tail via V_WMMA_F32_16X16X4_F32 ----------------
// one wave per 16 rows: out1 = relu(P[16x32] @ L1W[32x64] + b1) via WMMA,
// then out = relu(out1 @ L2W[64x2] + b2) scalar through LDS.
__global__ void k_mlp(const float* __restrict__ P, const float* __restrict__ W1g,
                      const float* __restrict__ b1g, const float* __restrict__ W2g,
                      const float* __restrict__ b2g, float* __restrict__ out, int Bg) {
    __shared__ float lds[16 * 64];
    int lane = threadIdx.x;
    int r0   = blockIdx.x * 16;
    int half = lane >> 4;      // which K half (A/B fragment layout)
    int l15  = lane & 15;
    int ar   = r0 + l15;       // A row this lane feeds
    bool av  = ar < Bg;

    #pragma unroll
    for (int nt = 0; nt < 4; ++nt) {
        int n0 = nt * 16;
        v8f acc = {0.f, 0.f, 0.f, 0.f, 0.f, 0.f, 0.f, 0.f};
        #pragma unroll
        for (int k0 = 0; k0 < 32; k0 += 4) {
            int ka = k0 + 2 * half;
            v2f af, bf;
            af.x = av ? P[(size_t)ar * 32 + ka]     : 0.0f;
            af.y = av ? P[(size_t)ar * 32 + ka + 1] : 0.0f;
            bf.x = W1g[ka * 64 + n0 + l15];
            bf.y = W1g[(ka + 1) * 64 + n0 + l15];
            acc = __builtin_amdgcn_wmma_f32_16x16x4_f32(false, af, false, bf,
                                                        (short)0, acc, false, false);
        }
        float bias = b1g[n0 + l15];
        #pragma unroll
        for (int r = 0; r < 8; ++r) {
            float v = acc[r] + bias;            // D: vgpr r -> row r (+8 for hi lanes)
            v = v > 0.0f ? v : 0.0f;
            lds[(r + 8 * half) * 64 + n0 + l15] = v;
        }
    }
    __syncthreads();
    int row = lane >> 1, oc = lane & 1;
    float acc2 = b2g[oc];
    #pragma unroll
    for (int k = 0; k < 64; ++k) acc2 += lds[row * 64 + k] * W2g[k * 2 + oc];
    acc2 = acc2 > 0.0f ? acc2 : 0.0f;
    if (r0 + row < Bg) out[(size_t)(r0 + row) * 2 + oc] = acc2;
}

static inline unsigned gblk(long long n, int b) { return (unsigned)((n + b - 1) / b); }

extern "C" void kernel_launch(void* const* d_in, const int* in_sizes, int n_in,
                              void* d_out, int out_size, void* d_ws, size_t ws_size,
                              hipStream_t stream) {
    (void)n_in; (void)ws_size;
    const float* x     = (const float*)d_in[0];
    const int*   ei    = (const int*)d_in[1];
    const int*   batch = (const int*)d_in[2];
    const float* bn0_g = (const float*)d_in[3];
    const float* bn0_b = (const float*)d_in[4];
    const float* W1    = (const float*)d_in[5];
    const float* bn1_g = (const float*)d_in[7];
    const float* bn1_b = (const float*)d_in[8];
    const float* W2    = (const float*)d_in[9];
    const float* bn2_g = (const float*)d_in[11];
    const float* bn2_b = (const float*)d_in[12];
    const float* W3    = (const float*)d_in[13];
    const float* bn3_g = (const float*)d_in[15];
    const float* bn3_b = (const float*)d_in[16];
    const float* L1W   = (const float*)d_in[17];
    const float* L1b   = (const float*)d_in[18];
    const float* L2W   = (const float*)d_in[19];
    const float* L2b   = (const float*)d_in[20];

    const int       N  = in_sizes[0];
    const long long E  = (long long)in_sizes[1] / 2;
    const int       Bg = out_size / 2;
    const int*      srcI = ei;
    const int*      dstI = ei + E;

    // workspace layout (32-float stride feature rows)
    char*  w      = (char*)d_ws;
    float* F      = (float*)w;                               // 32N floats (agg/z)
    float* H      = (float*)(w + (size_t)N * 32 * 4);        // 32N floats (h)
    float* dinv   = (float*)(w + (size_t)N * 64 * 4);        // N floats (deg -> dinv)
    float* S      = (float*)(w + (size_t)N * 65 * 4);        // 512 floats of stats
    float* pooled = S + 512;                                  // 32*Bg floats

    // stats slots: [sum, ss, a, s] per layer
    float *sum0=S+0,   *ss0=S+1,   *a0=S+2,   *sf0=S+3;      // a0/sf0 contiguous for k_h1
    float *sum1=S+8,   *ss1=S+16,  *a1=S+24,  *sf1=S+32;
    float *sum2=S+48,  *ss2=S+64,  *a2=S+80,  *sf2=S+96;
    float *sum3=S+112, *ss3=S+144, *a3=S+176, *sf3=S+208;

    const float invN = 1.0f / (float)N;
    const long long zcount = (long long)N + 512 + (long long)Bg * 32;

    // 0) zero deg/stats/pooled
    k_zero<<<gblk(zcount, 256), 256, 0, stream>>>(dinv, zcount);
    // 1) degree + dinv
    k_deg<<<gblk(E, 256), 256, 0, stream>>>(dstI, (unsigned*)dinv, E);
    k_dinv<<<gblk(N, 256), 256, 0, stream>>>(dinv, N);
    // 2) bn0 on x
    k_stats<1><<<1024, 256, 0, stream>>>(x, 1, N, sum0, ss0);
    k_finalize<<<1, 1, 0, stream>>>(sum0, ss0, bn0_g, bn0_b, a0, sf0, invN);
    // 3) layer 1 (1 -> 8)
    k_h1<<<gblk(N, 256), 256, 0, stream>>>(x, W1, dinv, a0, F, H, N);
    k_edge<8><<<gblk(E * 2, 256), 256, 0, stream>>>(srcI, dstI, dinv, H, F, E);
    k_stats<8><<<1024, 256, 0, stream>>>(F, 32, N, sum1, ss1);
    k_finalize<<<1, 8, 0, stream>>>(sum1, ss1, bn1_g, bn1_b, a1, sf1, invN);
    // 4) layer 2 (8 -> 16)
    k_transform<8, 16><<<gblk(N, 256), 256, 0, stream>>>(F, H, dinv, a1, sf1, W2, N);
    k_edge<16><<<gblk(E * 4, 256), 256, 0, stream>>>(srcI, dstI, dinv, H, F, E);
    k_stats<16><<<1024, 256, 0, stream>>>(F, 32, N, sum2, ss2);
    k_finalize<<<1, 16, 0, stream>>>(sum2, ss2, bn2_g, bn2_b, a2, sf2, invN);
    // 5) layer 3 (16 -> 32)
    k_transform<16, 32><<<gblk(N, 256), 256, 0, stream>>>(F, H, dinv, a2, sf2, W3, N);
    k_edge<32><<<gblk(E * 8, 256), 256, 0, stream>>>(srcI, dstI, dinv, H, F, E);
    k_stats<32><<<1024, 256, 0, stream>>>(F, 32, N, sum3, ss3);
    k_finalize<<<1, 32, 0, stream>>>(sum3, ss3, bn3_g, bn3_b, a3, sf3, invN);
    // 6) pool + MLP (WMMA f32 16x16x4)
    k_pool<<<gblk(N, 256), 256, 0, stream>>>(F, batch, a3, sf3, pooled, N);
    k_mlp<<<gblk(Bg, 16), 32, 0, stream>>>(pooled, L1W, L1b, L2W, L2b, (float*)d_out, Bg);
}